// GCN_22265110462988
// MI455X (gfx1250) — compile-verified
//
#include <hip/hip_runtime.h>

// GCN inference for MI455X (gfx1250, wave32).
// Bottleneck is edge aggregation (~2.7 GB of gather + fp32 atomics @ 23.3 TB/s);
// GEMMs (11.5 GFLOP) use native-precision V_WMMA_F32_16X16X4_F32.
// W is staged in LDS K-pair-interleaved so every B fragment is one ds_load_b64.

typedef float v2f __attribute__((ext_vector_type(2)));
typedef float v8f __attribute__((ext_vector_type(8)));

// Guaranteed hardware fp32 atomic add at device scope (no CAS-loop fallback).
__device__ __forceinline__ void global_fadd(float* p, float v) {
    asm volatile("global_atomic_add_f32 %0, %1, off scope:SCOPE_DEV"
                 :: "v"(p), "v"(v) : "memory");
}

// ---------------- degree / normalization ----------------
__global__ __launch_bounds__(256) void k_deg_init(float* deg, int n) {
    int i = blockIdx.x * 256 + threadIdx.x;
    if (i < n) deg[i] = 1.0f;               // self-loop
}

__global__ __launch_bounds__(256) void k_deg_count(const int* __restrict__ dst,
                                                   float* deg, int e) {
    int i = blockIdx.x * 256 + threadIdx.x;
    if (i < e) global_fadd(&deg[dst[i]], 1.0f);
}

__global__ __launch_bounds__(256) void k_rsqrt(float* deg, int n) {
    int i = blockIdx.x * 256 + threadIdx.x;
    if (i < n) deg[i] = rsqrtf(deg[i]);     // deg >= 1 always
}

// ---------------- WMMA GEMM with fused epilogue ----------------
// One wave computes a 16-row tile across all NCOLS columns; K = 128.
// B-matrix LDS layout: pair p = k>>1, element = sW[p*PSTR + col*2 + (k&1)].
//   -> lane fragment {W[k][col], W[k+1][col]} is one aligned ds_load_b64.
//   -> PSTR = 2*NCOLS + 32: pair p vs p+1 differ by 32 banks -> conflict-free.
// EPI==0: val = (A@W)*dinv[row], dual-store into U and Acc (Acc init == self-loop).
// EPI==1: val = sigmoid((A@W) + bias[col]) -> U (final output).
template <int NCOLS, int EPI>
__global__ __launch_bounds__(256) void gcn_wmma_gemm(
    const float* __restrict__ A, const float* __restrict__ W,
    const float* __restrict__ dinv, const float* __restrict__ bias,
    float* __restrict__ U, float* __restrict__ Acc, int M)
{
    constexpr int NT   = NCOLS / 16;
    constexpr int PSTR = 2 * NCOLS + 32;           // dwords per K-pair row
    __shared__ float sW[64 * PSTR];

    // Fill: thread handles (pair p, col c); two coalesced global reads, one b64 store.
    for (int i = threadIdx.x; i < 64 * NCOLS; i += 256) {
        int p = i / NCOLS, c = i % NCOLS;
        v2f w;
        w.x = W[(2 * p + 0) * NCOLS + c];
        w.y = W[(2 * p + 1) * NCOLS + c];
        *(v2f*)&sW[p * PSTR + 2 * c] = w;
    }
    __syncthreads();

    const int lane = threadIdx.x & 31;
    const int wid  = threadIdx.x >> 5;
    const int m0   = (blockIdx.x * 8 + wid) * 16;
    if (m0 >= M) return;

    const int r  = lane & 15;          // A row / B col within tile
    const int hi = lane >> 4;          // 0: K{0,1}, 1: K{2,3}

    int arow = m0 + r; if (arow >= M) arow = M - 1;
    const float* __restrict__ Ap = A + (size_t)arow * 128 + 2 * hi;
    const float* __restrict__ Bp = sW + hi * PSTR + 2 * r;

    v8f c[NT];
#pragma unroll
    for (int t = 0; t < NT; ++t)
#pragma unroll
        for (int j = 0; j < 8; ++j) c[t][j] = 0.0f;

    for (int k0 = 0; k0 < 128; k0 += 4) {
        const v2f a = *(const v2f*)(Ap + k0);                  // global_load_b64
        const float* brow = Bp + (k0 >> 1) * PSTR;
#pragma unroll
        for (int t = 0; t < NT; ++t) {
            const v2f b = *(const v2f*)(brow + t * 32);        // ds_load_b64
            c[t] = __builtin_amdgcn_wmma_f32_16x16x4_f32(
                false, a, false, b, (short)0, c[t], false, false);
        }
    }

    const int rhi = hi * 8;            // C/D: VGPR v -> row v (+8 for upper half-wave)
#pragma unroll
    for (int t = 0; t < NT; ++t) {
#pragma unroll
        for (int v = 0; v < 8; ++v) {
            int row = m0 + rhi + v;
            if (row >= M) continue;
            int col = t * 16 + r;
            size_t o = (size_t)row * NCOLS + col;
            float val = c[t][v];
            if (EPI == 0) {
                val *= dinv[row];
                U[o]   = val;
                Acc[o] = val;          // self-loop contribution
            } else {
                val += bias[col];
                U[o] = 1.0f / (1.0f + __expf(-val));
            }
        }
    }
}

// ---------------- edge aggregation: acc[dst] += u[src] ----------------
// One wave per edge: 32 lanes x float4 = 128 floats, coalesced b128 load
// + hardware fp32 atomics to consecutive addresses.
__global__ __launch_bounds__(256) void gcn_edge_agg(
    const int* __restrict__ src, const int* __restrict__ dst,
    const float* __restrict__ U, float* __restrict__ Acc, int E)
{
    int e = blockIdx.x * 8 + (threadIdx.x >> 5);
    if (e >= E) return;
    int lane = threadIdx.x & 31;
    int s = src[e];
    int d = dst[e];
    float4 val = ((const float4*)(U + (size_t)s * 128))[lane];
    float* p = Acc + (size_t)d * 128 + lane * 4;
    global_fadd(p + 0, val.x);
    global_fadd(p + 1, val.y);
    global_fadd(p + 2, val.z);
    global_fadd(p + 3, val.w);
}

// ---------------- finalize: h = relu(dinv[row]*acc + bias) ----------------
__global__ __launch_bounds__(256) void gcn_finalize(
    const float4* __restrict__ acc, const float* __restrict__ dinv,
    const float4* __restrict__ bias, float4* __restrict__ h, int n4)
{
    int i = blockIdx.x * 256 + threadIdx.x;
    if (i >= n4) return;
    float dv = dinv[i >> 5];           // 32 float4 per row of 128
    float4 a = acc[i];
    float4 b = bias[i & 31];
    float4 o;
    o.x = fmaxf(dv * a.x + b.x, 0.0f);
    o.y = fmaxf(dv * a.y + b.y, 0.0f);
    o.z = fmaxf(dv * a.z + b.z, 0.0f);
    o.w = fmaxf(dv * a.w + b.w, 0.0f);
    h[i] = o;
}

extern "C" void kernel_launch(void* const* d_in, const int* in_sizes, int n_in,
                              void* d_out, int out_size, void* d_ws, size_t ws_size,
                              hipStream_t stream)
{
    const float* x    = (const float*)d_in[0];
    const int*   ei   = (const int*)d_in[1];
    const float* W0   = (const float*)d_in[2];
    const float* b0   = (const float*)d_in[3];
    const float* W1   = (const float*)d_in[4];
    const float* b1   = (const float*)d_in[5];
    const float* W2   = (const float*)d_in[6];
    const float* b2   = (const float*)d_in[7];
    const float* Wout = (const float*)d_in[8];
    const float* bout = (const float*)d_in[9];

    const int N = in_sizes[0] / 128;
    const int E = in_sizes[1] / 2;
    const int* src = ei;
    const int* dst = ei + E;

    char* ws = (char*)d_ws;
    float* dinv = (float*)ws;
    size_t off = ((size_t)N * sizeof(float) + 255) & ~(size_t)255;
    float* bufU = (float*)(ws + off);
    off += (size_t)N * 128 * sizeof(float);
    float* bufAcc = (float*)(ws + off);

    // normalization: deg (incl. self-loops) -> 1/sqrt(deg)
    k_deg_init <<<(N + 255) / 256, 256, 0, stream>>>(dinv, N);
    k_deg_count<<<(E + 255) / 256, 256, 0, stream>>>(dst, dinv, E);
    k_rsqrt    <<<(N + 255) / 256, 256, 0, stream>>>(dinv, N);

    const int gemm_blocks = ((N + 15) / 16 + 7) / 8;
    const int agg_blocks  = (E + 7) / 8;
    const int fin_blocks  = (N * 32 + 255) / 256;

    // layer 0 (reads input x)
    gcn_wmma_gemm<128, 0><<<gemm_blocks, 256, 0, stream>>>(x, W0, dinv, nullptr, bufU, bufAcc, N);
    gcn_edge_agg<<<agg_blocks, 256, 0, stream>>>(src, dst, bufU, bufAcc, E);
    gcn_finalize<<<fin_blocks, 256, 0, stream>>>((const float4*)bufAcc, dinv, (const float4*)b0, (float4*)bufU, N * 32);
    // layer 1 (in-place on bufU: each wave owns its 16 rows exclusively)
    gcn_wmma_gemm<128, 0><<<gemm_blocks, 256, 0, stream>>>(bufU, W1, dinv, nullptr, bufU, bufAcc, N);
    gcn_edge_agg<<<agg_blocks, 256, 0, stream>>>(src, dst, bufU, bufAcc, E);
    gcn_finalize<<<fin_blocks, 256, 0, stream>>>((const float4*)bufAcc, dinv, (const float4*)b1, (float4*)bufU, N * 32);
    // layer 2
    gcn_wmma_gemm<128, 0><<<gemm_blocks, 256, 0, stream>>>(bufU, W2, dinv, nullptr, bufU, bufAcc, N);
    gcn_edge_agg<<<agg_blocks, 256, 0, stream>>>(src, dst, bufU, bufAcc, E);
    gcn_finalize<<<fin_blocks, 256, 0, stream>>>((const float4*)bufAcc, dinv, (const float4*)b2, (float4*)bufU, N * 32);
    // output layer: sigmoid(h @ W_out + b_out) -> d_out
    gcn_wmma_gemm<64, 1><<<gemm_blocks, 256, 0, stream>>>(bufU, Wout, dinv, bout, (float*)d_out, nullptr, N);
}